// EGNNLayer_7653631721502
// MI455X (gfx1250) — compile-verified
//
#include <hip/hip_runtime.h>

// ---------------------------------------------------------------------------
// E(3)-equivariant GNN layer for MI455X (gfx1250), fp32 WMMA (16x16x4).
// Tensor products expressed as [tile x K] x [K x N] GEMMs over outer-product
// features built on the fly from LDS.  Weights are repacked per-launch into a
// [K/2][N][2] "paired" layout so each lane's two B values per WMMA are one
// global_load_b64; the message kernel runs M=32 per wave so each B load feeds
// two WMMAs (vmem:wmma issue ratio 1:2 in the hot loop).
// ---------------------------------------------------------------------------

typedef float v2f __attribute__((ext_vector_type(2)));
typedef float v8f __attribute__((ext_vector_type(8)));

#define N_NODES 10000
#define N_EDGES 100000

// normalization constants folded into A-operand features
#define SCZ_MSG 0.03608439182435161f   // 1/sqrt(768)
#define SCD_MSG 0.02083333333333333f   // 1/sqrt(3*768) = 1/48
#define SC_LIN_S 0.125f                // 1/sqrt(64)
#define SC_LIN_V 0.17677669529663687f  // 1/sqrt(32)
#define SCZ_UPD 0.011811347247226025f  // 1/sqrt(7168)
#define SCD_UPD 0.006819336486389271f  // 1/sqrt(3*7168)
#define SCV_UPD 0.013975424859373686f  // 1/sqrt(5120)

__device__ __forceinline__ v8f vzero8() {
  v8f z = {0.f, 0.f, 0.f, 0.f, 0.f, 0.f, 0.f, 0.f};
  return z;
}

// D = A(16x4)*B(4x16) + C, fp32.  Lane mapping: A/B element r has
// K = 2*(lane>>4)+r; A row / B col = lane&15; C/D elem r: row r+8*(lane>>4).
__device__ __forceinline__ v8f wmma4(float a0, float a1, v2f b, v8f c) {
  v2f a = {a0, a1};
  return __builtin_amdgcn_wmma_f32_16x16x4_f32(false, a, false, b, (short)0, c,
                                               false, false);
}

// ---------------------------------------------------------------------------
// Kernel: zero the aggregation workspace
// ---------------------------------------------------------------------------
__global__ void egnn_zero_kernel(float* __restrict__ p, int n) {
  int i = blockIdx.x * blockDim.x + threadIdx.x;
  int stride = gridDim.x * blockDim.x;
  for (; i < n; i += stride) p[i] = 0.0f;
}

// ---------------------------------------------------------------------------
// Kernel: repack W[K][N] -> P[K/2][N][2] so (W[2kp][n], W[2kp+1][n]) is one
// contiguous float2 (global_load_b64 in the GEMM loops).
// ---------------------------------------------------------------------------
__global__ void egnn_pack_kernel(const float* __restrict__ src,
                                 float* __restrict__ dst, int K, int N) {
  int i = blockIdx.x * blockDim.x + threadIdx.x;
  int total = K * N;
  int stride = gridDim.x * blockDim.x;
  for (; i < total; i += stride) {
    int k = i / N, n = i - k * N;
    dst[((k >> 1) * N + n) * 2 + (k & 1)] = src[i];
  }
}

// ---------------------------------------------------------------------------
// Kernel 1: edge messages.  One wave = 32 edges (2 sub-tiles of 16 -> every
// B load is shared by 2 WMMAs).
//   gather -> TP scalars [32x768]@[768x96] -> gate
//          -> linear [32x64]@[64x96] -> scatter
//          -> TP vectors [32x768]@[768x32] per m -> gate
//          -> linear [32x32]@[32x32] per m -> scatter
// agg layout per node: [0..95]=agg_s, [96+w*3+m]=agg_v[w][m]
// ---------------------------------------------------------------------------
__global__ __launch_bounds__(32) void egnn_msg_kernel(
    const float* __restrict__ nodef, const float* __restrict__ edgef,
    const float* __restrict__ pw_ss, const float* __restrict__ pw_vv,
    const float* __restrict__ pw_sv, const float* __restrict__ pw_vs,
    const float* __restrict__ pw_lin_s, const float* __restrict__ pw_lin_v,
    const int* __restrict__ eidx, float* __restrict__ agg) {
  __shared__ float sXS[2][16][64];   // src node scalars
  __shared__ float sXV[2][16][96];   // src node vectors (u*3+m)
  __shared__ float sESC[2][16][8];   // edge scalars
  __shared__ float sEVC[2][16][24];  // edge vectors (v*3+m)
  __shared__ float sMS[2][16][96];   // scalar messages -> silu/sigmoid
  __shared__ float sPV[2][16][96];   // gated vectors (m*32+w)
  __shared__ int sROW[32];
  __shared__ int sCOL[32];

  const v2f* __restrict__ pss = (const v2f*)pw_ss;
  const v2f* __restrict__ pvv = (const v2f*)pw_vv;
  const v2f* __restrict__ psv = (const v2f*)pw_sv;
  const v2f* __restrict__ pvs = (const v2f*)pw_vs;
  const v2f* __restrict__ pls = (const v2f*)pw_lin_s;
  const v2f* __restrict__ plv = (const v2f*)pw_lin_v;

  const int lane = threadIdx.x & 31;
  const int hlf = lane >> 4;
  const int lc = lane & 15;
  const long ebase = (long)blockIdx.x * 32;

  {
    long e = ebase + lane;
    if (e >= N_EDGES) e = N_EDGES - 1;
    sROW[lane] = eidx[e];            // edge_index[0][e] : destination
    sCOL[lane] = eidx[N_EDGES + e];  // edge_index[1][e] : source
  }
  __syncthreads();

  // gather source-node features (all nodes = 6.4MB, L2-resident)
  for (int idx = lane; idx < 32 * 160; idx += 32) {
    int e = idx / 160, f = idx - e * 160;
    int s = e >> 4, el = e & 15;
    long j = sCOL[e];
    float v = nodef[j * 160 + f];
    if (f < 64) sXS[s][el][f] = v;
    else        sXV[s][el][f - 64] = v;
  }
  for (int idx = lane; idx < 32 * 32; idx += 32) {
    int e = idx / 32, f = idx - e * 32;
    int s = e >> 4, el = e & 15;
    long ee = ebase + e;
    if (ee >= N_EDGES) ee = N_EDGES - 1;
    float v = edgef[ee * 32 + f];
    if (f < 8) sESC[s][el][f] = v;
    else       sEVC[s][el][f - 8] = v;
  }
  __syncthreads();

  // ---- TP scalar outputs: K=768 (512 ss + 256 vv), N=96 ----
  v8f acc[2][6];
#pragma unroll
  for (int s = 0; s < 2; ++s)
#pragma unroll
    for (int t = 0; t < 6; ++t) acc[s][t] = vzero8();

#pragma unroll 4
  for (int kk = 0; kk < 128; ++kk) {  // ss path: k = u*8+v
    const int k0 = kk * 4 + hlf * 2, k1 = k0 + 1;
    const int kp = kk * 2 + hlf;
    float aa[2][2];
#pragma unroll
    for (int s = 0; s < 2; ++s) {
      aa[s][0] = sXS[s][lc][k0 >> 3] * sESC[s][lc][k0 & 7] * SCZ_MSG;
      aa[s][1] = sXS[s][lc][k1 >> 3] * sESC[s][lc][k1 & 7] * SCZ_MSG;
    }
#pragma unroll
    for (int t = 0; t < 6; ++t) {
      const v2f b = pss[kp * 96 + t * 16 + lc];
#pragma unroll
      for (int s = 0; s < 2; ++s) acc[s][t] = wmma4(aa[s][0], aa[s][1], b, acc[s][t]);
    }
  }
#pragma unroll 4
  for (int kk = 0; kk < 64; ++kk) {  // vv path: dot over m, k = u*8+v
    const int k0 = kk * 4 + hlf * 2, k1 = k0 + 1;
    const int kp = kk * 2 + hlf;
    const int u0 = k0 >> 3, v0 = k0 & 7, u1 = k1 >> 3, v1 = k1 & 7;
    float aa[2][2];
#pragma unroll
    for (int s = 0; s < 2; ++s) {
      aa[s][0] = (sXV[s][lc][u0 * 3 + 0] * sEVC[s][lc][v0 * 3 + 0] +
                  sXV[s][lc][u0 * 3 + 1] * sEVC[s][lc][v0 * 3 + 1] +
                  sXV[s][lc][u0 * 3 + 2] * sEVC[s][lc][v0 * 3 + 2]) * SCD_MSG;
      aa[s][1] = (sXV[s][lc][u1 * 3 + 0] * sEVC[s][lc][v1 * 3 + 0] +
                  sXV[s][lc][u1 * 3 + 1] * sEVC[s][lc][v1 * 3 + 1] +
                  sXV[s][lc][u1 * 3 + 2] * sEVC[s][lc][v1 * 3 + 2]) * SCD_MSG;
    }
#pragma unroll
    for (int t = 0; t < 6; ++t) {
      const v2f b = pvv[kp * 96 + t * 16 + lc];
#pragma unroll
      for (int s = 0; s < 2; ++s) acc[s][t] = wmma4(aa[s][0], aa[s][1], b, acc[s][t]);
    }
  }

#pragma unroll
  for (int s = 0; s < 2; ++s)
#pragma unroll
    for (int t = 0; t < 6; ++t)
#pragma unroll
      for (int r = 0; r < 8; ++r)
        sMS[s][r + hlf * 8][t * 16 + lc] = acc[s][t][r];
  __syncthreads();

  // gate: silu on [0,64), sigmoid gates on [64,96)
  for (int idx = lane; idx < 32 * 96; idx += 32) {
    int e = idx / 96, w = idx - e * 96;
    int s = e >> 4, el = e & 15;
    float x = sMS[s][el][w];
    float sg = 1.0f / (1.0f + __expf(-x));
    sMS[s][el][w] = (w < 64) ? x * sg : sg;
  }
  __syncthreads();

  // ---- linear scalars [32x64]@[64x96], then scatter-sum ----
#pragma unroll
  for (int s = 0; s < 2; ++s)
#pragma unroll
    for (int t = 0; t < 6; ++t) acc[s][t] = vzero8();
#pragma unroll 4
  for (int kk = 0; kk < 16; ++kk) {
    const int k0 = kk * 4 + hlf * 2, k1 = k0 + 1;
    const int kp = kk * 2 + hlf;
    float aa[2][2];
#pragma unroll
    for (int s = 0; s < 2; ++s) {
      aa[s][0] = sMS[s][lc][k0] * SC_LIN_S;
      aa[s][1] = sMS[s][lc][k1] * SC_LIN_S;
    }
#pragma unroll
    for (int t = 0; t < 6; ++t) {
      const v2f b = pls[kp * 96 + t * 16 + lc];
#pragma unroll
      for (int s = 0; s < 2; ++s) acc[s][t] = wmma4(aa[s][0], aa[s][1], b, acc[s][t]);
    }
  }
#pragma unroll
  for (int s = 0; s < 2; ++s)
#pragma unroll
    for (int t = 0; t < 6; ++t)
#pragma unroll
      for (int r = 0; r < 8; ++r) {
        const int er = r + hlf * 8;
        if (ebase + s * 16 + er < N_EDGES)
          atomicAdd(&agg[(long)sROW[s * 16 + er] * 192 + t * 16 + lc], acc[s][t][r]);
      }

  // ---- TP vector outputs per m: K=768 (512 sv + 256 vs), N=32 ----
#pragma unroll 1
  for (int m = 0; m < 3; ++m) {
    v8f vac[2][2];
#pragma unroll
    for (int s = 0; s < 2; ++s) { vac[s][0] = vzero8(); vac[s][1] = vzero8(); }
#pragma unroll 4
    for (int kk = 0; kk < 128; ++kk) {  // sv: xs[u]*yv[v][m]
      const int k0 = kk * 4 + hlf * 2, k1 = k0 + 1;
      const int kp = kk * 2 + hlf;
      float aa[2][2];
#pragma unroll
      for (int s = 0; s < 2; ++s) {
        aa[s][0] = sXS[s][lc][k0 >> 3] * sEVC[s][lc][(k0 & 7) * 3 + m] * SCZ_MSG;
        aa[s][1] = sXS[s][lc][k1 >> 3] * sEVC[s][lc][(k1 & 7) * 3 + m] * SCZ_MSG;
      }
#pragma unroll
      for (int t = 0; t < 2; ++t) {
        const v2f b = psv[kp * 32 + t * 16 + lc];
#pragma unroll
        for (int s = 0; s < 2; ++s) vac[s][t] = wmma4(aa[s][0], aa[s][1], b, vac[s][t]);
      }
    }
#pragma unroll 4
    for (int kk = 0; kk < 64; ++kk) {  // vs: xv[u][m]*ys[v]
      const int k0 = kk * 4 + hlf * 2, k1 = k0 + 1;
      const int kp = kk * 2 + hlf;
      float aa[2][2];
#pragma unroll
      for (int s = 0; s < 2; ++s) {
        aa[s][0] = sXV[s][lc][(k0 >> 3) * 3 + m] * sESC[s][lc][k0 & 7] * SCZ_MSG;
        aa[s][1] = sXV[s][lc][(k1 >> 3) * 3 + m] * sESC[s][lc][k1 & 7] * SCZ_MSG;
      }
#pragma unroll
      for (int t = 0; t < 2; ++t) {
        const v2f b = pvs[kp * 32 + t * 16 + lc];
#pragma unroll
        for (int s = 0; s < 2; ++s) vac[s][t] = wmma4(aa[s][0], aa[s][1], b, vac[s][t]);
      }
    }
    // gate vectors (gates live at sMS[.][64+w]) and stash for the linear
#pragma unroll
    for (int s = 0; s < 2; ++s)
#pragma unroll
      for (int t = 0; t < 2; ++t)
#pragma unroll
        for (int r = 0; r < 8; ++r) {
          const int er = r + hlf * 8;
          const int w = t * 16 + lc;
          sPV[s][er][m * 32 + w] = vac[s][t][r] * sMS[s][er][64 + w];
        }
  }
  __syncthreads();

  // ---- linear vectors [32x32]@[32x32] per m, then scatter-sum ----
#pragma unroll 1
  for (int m = 0; m < 3; ++m) {
    v8f vac[2][2];
#pragma unroll
    for (int s = 0; s < 2; ++s) { vac[s][0] = vzero8(); vac[s][1] = vzero8(); }
#pragma unroll
    for (int kk = 0; kk < 8; ++kk) {
      const int k0 = kk * 4 + hlf * 2, k1 = k0 + 1;
      const int kp = kk * 2 + hlf;
      float aa[2][2];
#pragma unroll
      for (int s = 0; s < 2; ++s) {
        aa[s][0] = sPV[s][lc][m * 32 + k0] * SC_LIN_V;
        aa[s][1] = sPV[s][lc][m * 32 + k1] * SC_LIN_V;
      }
#pragma unroll
      for (int t = 0; t < 2; ++t) {
        const v2f b = plv[kp * 32 + t * 16 + lc];
#pragma unroll
        for (int s = 0; s < 2; ++s) vac[s][t] = wmma4(aa[s][0], aa[s][1], b, vac[s][t]);
      }
    }
#pragma unroll
    for (int s = 0; s < 2; ++s)
#pragma unroll
      for (int t = 0; t < 2; ++t)
#pragma unroll
        for (int r = 0; r < 8; ++r) {
          const int er = r + hlf * 8;
          if (ebase + s * 16 + er < N_EDGES)
            atomicAdd(&agg[(long)sROW[s * 16 + er] * 192 + 96 + (t * 16 + lc) * 3 + m],
                      vac[s][t][r]);
        }
  }
}

// ---------------------------------------------------------------------------
// Kernel 2: node update.  One wave = one 16-node tile, paired B loads.
// ---------------------------------------------------------------------------
__global__ __launch_bounds__(32) void egnn_upd_kernel(
    const float* __restrict__ nodef, const float* __restrict__ pw_ss,
    const float* __restrict__ pw_vv, const float* __restrict__ pw_sv,
    const float* __restrict__ pw_vs, const float* __restrict__ pw_lin_s,
    const float* __restrict__ pw_lin_v, const float* __restrict__ agg,
    float* __restrict__ out) {
  __shared__ float sNS[16][64];
  __shared__ float sNV[16][96];
  __shared__ float sAS[16][96];
  __shared__ float sAV[16][96];
  __shared__ float sMS[16][96];
  __shared__ float sPV[16][96];

  const v2f* __restrict__ pss = (const v2f*)pw_ss;
  const v2f* __restrict__ pvv = (const v2f*)pw_vv;
  const v2f* __restrict__ psv = (const v2f*)pw_sv;
  const v2f* __restrict__ pvs = (const v2f*)pw_vs;
  const v2f* __restrict__ pls = (const v2f*)pw_lin_s;
  const v2f* __restrict__ plv = (const v2f*)pw_lin_v;

  const int lane = threadIdx.x & 31;
  const int hlf = lane >> 4;
  const int lc = lane & 15;
  const long nbase = (long)blockIdx.x * 16;

  for (int idx = lane; idx < 16 * 160; idx += 32) {
    int e = idx / 160, f = idx - e * 160;
    long g = nbase + e;
    if (g >= N_NODES) g = N_NODES - 1;
    float v = nodef[g * 160 + f];
    if (f < 64) sNS[e][f] = v;
    else        sNV[e][f - 64] = v;
  }
  for (int idx = lane; idx < 16 * 192; idx += 32) {
    int e = idx / 192, f = idx - e * 192;
    long g = nbase + e;
    if (g >= N_NODES) g = N_NODES - 1;
    float v = agg[g * 192 + f];
    if (f < 96) sAS[e][f] = v;
    else        sAV[e][f - 96] = v;
  }
  __syncthreads();

  // ---- TP scalars: K=7168 (6144 ss + 1024 vv), N=96 ----
  v8f acc[6];
#pragma unroll
  for (int t = 0; t < 6; ++t) acc[t] = vzero8();

#pragma unroll 2
  for (int kk = 0; kk < 1536; ++kk) {  // ss: k = u*96+v (v even -> v1=v+1)
    const int k0 = kk * 4 + hlf * 2;
    const int kp = kk * 2 + hlf;
    const int u0 = k0 / 96;
    const int v0 = k0 - u0 * 96;
    const float xs = sNS[lc][u0];
    const float a0 = xs * sAS[lc][v0] * SCZ_UPD;
    const float a1 = xs * sAS[lc][v0 + 1] * SCZ_UPD;
#pragma unroll
    for (int t = 0; t < 6; ++t)
      acc[t] = wmma4(a0, a1, pss[kp * 96 + t * 16 + lc], acc[t]);
  }
#pragma unroll 2
  for (int kk = 0; kk < 256; ++kk) {  // vv: k = u*32+v, dot over m
    const int k0 = kk * 4 + hlf * 2;
    const int kp = kk * 2 + hlf;
    const int u0 = k0 >> 5;
    const int v0 = k0 & 31;
    const float x0 = sNV[lc][u0 * 3 + 0];
    const float x1 = sNV[lc][u0 * 3 + 1];
    const float x2 = sNV[lc][u0 * 3 + 2];
    const float a0 = (x0 * sAV[lc][v0 * 3 + 0] + x1 * sAV[lc][v0 * 3 + 1] +
                      x2 * sAV[lc][v0 * 3 + 2]) * SCD_UPD;
    const float a1 = (x0 * sAV[lc][(v0 + 1) * 3 + 0] + x1 * sAV[lc][(v0 + 1) * 3 + 1] +
                      x2 * sAV[lc][(v0 + 1) * 3 + 2]) * SCD_UPD;
#pragma unroll
    for (int t = 0; t < 6; ++t)
      acc[t] = wmma4(a0, a1, pvv[kp * 96 + t * 16 + lc], acc[t]);
  }

#pragma unroll
  for (int t = 0; t < 6; ++t)
#pragma unroll
    for (int r = 0; r < 8; ++r)
      sMS[r + hlf * 8][t * 16 + lc] = acc[t][r];
  __syncthreads();

  for (int idx = lane; idx < 16 * 96; idx += 32) {
    int e = idx / 96, w = idx - e * 96;
    float x = sMS[e][w];
    float sg = 1.0f / (1.0f + __expf(-x));
    sMS[e][w] = (w < 64) ? x * sg : sg;
  }
  __syncthreads();

  // ---- TP vectors per m: K=5120 (2048 sv + 3072 vs), N=32 ----
#pragma unroll 1
  for (int m = 0; m < 3; ++m) {
    v8f vac[2];
    vac[0] = vzero8();
    vac[1] = vzero8();
#pragma unroll 2
    for (int kk = 0; kk < 512; ++kk) {  // sv: k = u*32+v, ns[u]*agg_v[v][m]
      const int k0 = kk * 4 + hlf * 2;
      const int kp = kk * 2 + hlf;
      const int u0 = k0 >> 5;
      const int v0 = k0 & 31;
      const float xs = sNS[lc][u0];
      const float a0 = xs * sAV[lc][v0 * 3 + m] * SCV_UPD;
      const float a1 = xs * sAV[lc][(v0 + 1) * 3 + m] * SCV_UPD;
#pragma unroll
      for (int t = 0; t < 2; ++t)
        vac[t] = wmma4(a0, a1, psv[kp * 32 + t * 16 + lc], vac[t]);
    }
#pragma unroll 2
    for (int kk = 0; kk < 768; ++kk) {  // vs: k = u*96+v, nv[u][m]*agg_s[v]
      const int k0 = kk * 4 + hlf * 2;
      const int kp = kk * 2 + hlf;
      const int u0 = k0 / 96;
      const int v0 = k0 - u0 * 96;
      const float xv = sNV[lc][u0 * 3 + m];
      const float a0 = xv * sAS[lc][v0] * SCV_UPD;
      const float a1 = xv * sAS[lc][v0 + 1] * SCV_UPD;
#pragma unroll
      for (int t = 0; t < 2; ++t)
        vac[t] = wmma4(a0, a1, pvs[kp * 32 + t * 16 + lc], vac[t]);
    }
#pragma unroll
    for (int t = 0; t < 2; ++t)
#pragma unroll
      for (int r = 0; r < 8; ++r) {
        const int er = r + hlf * 8;
        const int w = t * 16 + lc;
        sPV[er][m * 32 + w] = vac[t][r] * sMS[er][64 + w];
      }
  }
  __syncthreads();

  // ---- linear scalars [16x64]@[64x64] + residual ----
  v8f dac[4];
#pragma unroll
  for (int t = 0; t < 4; ++t) dac[t] = vzero8();
#pragma unroll 4
  for (int kk = 0; kk < 16; ++kk) {
    const int k0 = kk * 4 + hlf * 2, k1 = k0 + 1;
    const int kp = kk * 2 + hlf;
    const float a0 = sMS[lc][k0] * SC_LIN_S;
    const float a1 = sMS[lc][k1] * SC_LIN_S;
#pragma unroll
    for (int t = 0; t < 4; ++t)
      dac[t] = wmma4(a0, a1, pls[kp * 64 + t * 16 + lc], dac[t]);
  }
#pragma unroll
  for (int t = 0; t < 4; ++t)
#pragma unroll
    for (int r = 0; r < 8; ++r) {
      const int er = r + hlf * 8;
      const long g = nbase + er;
      if (g < N_NODES) {
        const int f = t * 16 + lc;
        out[g * 160 + f] = nodef[g * 160 + f] + dac[t][r];
      }
    }

  // ---- linear vectors [16x32]@[32x32] per m + residual ----
#pragma unroll 1
  for (int m = 0; m < 3; ++m) {
    v8f vac[2];
    vac[0] = vzero8();
    vac[1] = vzero8();
#pragma unroll
    for (int kk = 0; kk < 8; ++kk) {
      const int k0 = kk * 4 + hlf * 2, k1 = k0 + 1;
      const int kp = kk * 2 + hlf;
      const float a0 = sPV[lc][m * 32 + k0] * SC_LIN_V;
      const float a1 = sPV[lc][m * 32 + k1] * SC_LIN_V;
#pragma unroll
      for (int t = 0; t < 2; ++t)
        vac[t] = wmma4(a0, a1, plv[kp * 32 + t * 16 + lc], vac[t]);
    }
#pragma unroll
    for (int t = 0; t < 2; ++t)
#pragma unroll
      for (int r = 0; r < 8; ++r) {
        const int er = r + hlf * 8;
        const long g = nbase + er;
        if (g < N_NODES) {
          const int f = 64 + (t * 16 + lc) * 3 + m;
          out[g * 160 + f] = nodef[g * 160 + f] + vac[t][r];
        }
      }
  }
}

// ---------------------------------------------------------------------------
extern "C" void kernel_launch(void* const* d_in, const int* in_sizes, int n_in,
                              void* d_out, int out_size, void* d_ws, size_t ws_size,
                              hipStream_t stream) {
  (void)in_sizes; (void)n_in; (void)out_size; (void)ws_size;
  const float* nodef       = (const float*)d_in[0];
  const float* edgef       = (const float*)d_in[1];
  const float* w_msg_ss    = (const float*)d_in[2];
  const float* w_msg_vv    = (const float*)d_in[3];
  const float* w_msg_sv    = (const float*)d_in[4];
  const float* w_msg_vs    = (const float*)d_in[5];
  const float* w_lin_msg_s = (const float*)d_in[6];
  const float* w_lin_msg_v = (const float*)d_in[7];
  const float* w_upd_ss    = (const float*)d_in[8];
  const float* w_upd_vv    = (const float*)d_in[9];
  const float* w_upd_sv    = (const float*)d_in[10];
  const float* w_upd_vs    = (const float*)d_in[11];
  const float* w_lin_upd_s = (const float*)d_in[12];
  const float* w_lin_upd_v = (const float*)d_in[13];
  const int*   eidx        = (const int*)d_in[14];
  float* out = (float*)d_out;

  // workspace layout (floats): agg [N*192], then paired weight copies
  float* ws = (float*)d_ws;
  float* agg = ws;
  float* p = ws + (size_t)N_NODES * 192;
  float* p_msg_ss    = p; p += 512 * 96;
  float* p_msg_vv    = p; p += 256 * 96;
  float* p_msg_sv    = p; p += 512 * 32;
  float* p_msg_vs    = p; p += 256 * 32;
  float* p_lin_msg_s = p; p += 64 * 96;
  float* p_lin_msg_v = p; p += 32 * 32;
  float* p_upd_ss    = p; p += 6144 * 96;
  float* p_upd_vv    = p; p += 1024 * 96;
  float* p_upd_sv    = p; p += 2048 * 32;
  float* p_upd_vs    = p; p += 3072 * 32;
  float* p_lin_upd_s = p; p += 64 * 64;
  float* p_lin_upd_v = p; p += 32 * 32;

  egnn_zero_kernel<<<512, 256, 0, stream>>>(agg, N_NODES * 192);

  struct PackJob { const float* s; float* d; int K, N; };
  const PackJob jobs[12] = {
      {w_msg_ss, p_msg_ss, 512, 96},    {w_msg_vv, p_msg_vv, 256, 96},
      {w_msg_sv, p_msg_sv, 512, 32},    {w_msg_vs, p_msg_vs, 256, 32},
      {w_lin_msg_s, p_lin_msg_s, 64, 96}, {w_lin_msg_v, p_lin_msg_v, 32, 32},
      {w_upd_ss, p_upd_ss, 6144, 96},   {w_upd_vv, p_upd_vv, 1024, 96},
      {w_upd_sv, p_upd_sv, 2048, 32},   {w_upd_vs, p_upd_vs, 3072, 32},
      {w_lin_upd_s, p_lin_upd_s, 64, 64}, {w_lin_upd_v, p_lin_upd_v, 32, 32}};
  for (int i = 0; i < 12; ++i) {
    int total = jobs[i].K * jobs[i].N;
    int blocks = (total + 255) / 256;
    if (blocks > 512) blocks = 512;
    egnn_pack_kernel<<<blocks, 256, 0, stream>>>(jobs[i].s, jobs[i].d, jobs[i].K,
                                                 jobs[i].N);
  }

  const int blocks1 = (N_EDGES + 31) / 32;  // one wave = 32 edges
  egnn_msg_kernel<<<blocks1, 32, 0, stream>>>(
      nodef, edgef, p_msg_ss, p_msg_vv, p_msg_sv, p_msg_vs, p_lin_msg_s,
      p_lin_msg_v, eidx, agg);

  const int ntiles = (N_NODES + 15) / 16;
  egnn_upd_kernel<<<ntiles, 32, 0, stream>>>(nodef, p_upd_ss, p_upd_vv, p_upd_sv,
                                             p_upd_vs, p_lin_upd_s, p_lin_upd_v,
                                             agg, out);
}